// CH_37495064494410
// MI455X (gfx1250) — compile-verified
//
#include <hip/hip_runtime.h>

// Cahn-Hilliard explicit step, fully fused double-stencil.
//   z  = lap(x)            (5-pt, edge-replicated)
//   x1 = -z - x + x^3
//   out = x + 0.1*lap(x1)  (5-pt, edge-replicated on x1)
//
// Memory-bound: ~268MB min traffic -> ~11.5us floor @ 23.3 TB/s.
// One pass, LDS tile. Interior tiles staged by the Tensor Data Mover
// (one tensor_load_to_lds per block); border tiles via async b32 loads
// with edge clamping. b128 global stores.

#define TX 64
#define TY 64
#define HALOY 2
#define XOFF 4                  // x halo widened to 4 so rows are 16B aligned
#define AWS (TX + 2 * XOFF)     // 72 floats per LDS row (16B-aligned segments)
#define AH (TY + 2 * HALOY)     // 68 rows
#define NSEG (AWS / 4)          // 18 float4 segments per row
#define BW (TX + 2)             // 66
#define BH (TY + 2)             // 66
#define OSEG (TX / 4)           // 16 float4 outputs per row
#define NTHREADS 256

// ---- CDNA5 feature detection (all guarded with plain-load fallbacks) ----
#if defined(__has_builtin)
#if __has_builtin(__builtin_amdgcn_global_load_async_to_lds_b32)
#define CH_ASYNC32 1
#endif
#if __has_builtin(__builtin_amdgcn_global_load_async_to_lds_b128)
#define CH_ASYNC128 1
#endif
#if __has_builtin(__builtin_amdgcn_tensor_load_to_lds)
#define CH_TDM 1
#endif
#endif

#if defined(CH_ASYNC32) || defined(CH_ASYNC128)
#define CH_ANY_ASYNC 1
#endif

typedef __attribute__((address_space(1))) int g1_int_t; // global (AS1) int
typedef __attribute__((address_space(3))) int l3_int_t; // LDS (AS3) int

#if defined(CH_ASYNC128)
typedef int v4i_t __attribute__((vector_size(4 * sizeof(int))));
typedef __attribute__((address_space(1))) v4i_t g1_v4i_t; // global (AS1) v4i
typedef __attribute__((address_space(3))) v4i_t l3_v4i_t; // LDS (AS3) v4i
#endif

#if defined(CH_TDM)
typedef unsigned int u32x4_t __attribute__((ext_vector_type(4)));
typedef int i32x8_t __attribute__((ext_vector_type(8)));
typedef int i32x4_t __attribute__((ext_vector_type(4)));
#endif

static __device__ __forceinline__ void wait_async_lds() {
#if defined(CH_ANY_ASYNC)
#if defined(__has_builtin)
#if __has_builtin(__builtin_amdgcn_s_wait_asynccnt)
  __builtin_amdgcn_s_wait_asynccnt(0);
#else
  asm volatile("s_wait_asynccnt 0" ::: "memory");
#endif
#else
  asm volatile("s_wait_asynccnt 0" ::: "memory");
#endif
#endif
}

#if defined(CH_TDM)
static __device__ __forceinline__ void wait_tensor() {
#if __has_builtin(__builtin_amdgcn_s_wait_tensorcnt)
  __builtin_amdgcn_s_wait_tensorcnt(0);
#else
  asm volatile("s_wait_tensorcnt 0" ::: "memory");
#endif
}
#endif

__global__ __launch_bounds__(NTHREADS) void ch_step_fused(
    const float* __restrict__ x, float* __restrict__ out, int W, int H) {
  __shared__ __align__(16) float a[AH * AWS]; // x tile, edge-clamped
  __shared__ __align__(16) float b[BH * BW];  // x1 tile, halo 1

  const int bx = blockIdx.x * TX;
  const int by = blockIdx.y * TY;
  const size_t img_off = (size_t)blockIdx.z * (size_t)W * (size_t)H;
  const float* __restrict__ img = x + img_off;
  float* __restrict__ oimg = out + img_off;

  const int tid = threadIdx.x;

  // ---- Stage 1: stage x-tile (rows [by-2,by+66), cols [bx-4,bx+68)) ----
  const bool interior =
      (bx >= XOFF) && (bx + TX + XOFF <= W) && (by >= HALOY) &&
      (by + TY + HALOY <= H); // block-uniform
  if (interior) {
#if defined(CH_TDM)
    // One Tensor-DMA per block: 68-row x 288B tile, row stride W*4 bytes,
    // packed contiguously into LDS (matches a[] layout exactly).
    if (tid < 32) { // wave 0 only (TDM ignores EXEC; gate by branch)
      const float* gp =
          img + (size_t)(by - HALOY) * (size_t)W + (size_t)(bx - XOFF);
      const unsigned long long ga = (unsigned long long)(size_t)gp;
      const unsigned lds_base = (unsigned)(size_t)(l3_int_t*)&a[0];

      u32x4_t g0;
      g0.x = 1u;                 // count=1 valid descriptor
      g0.y = lds_base;           // lds_addr (bytes)
      g0.z = (unsigned)ga;       // global_addr[31:0]
      g0.w = (unsigned)(ga >> 32) | 0x80000000u; // addr[56:32] | type=2<<30

      i32x8_t g1;
      g1.s0 = (int)(2u << 16);            // data_size=2 (4B), no mask/pad
      g1.s1 = (int)((unsigned)AWS << 16); // tensor_dim0[15:0] @ bits63:48
      g1.s2 = (int)((unsigned)AH << 16);  // td0[31:16]=0 | tensor_dim1[15:0]
      g1.s3 = (int)((unsigned)AWS << 16); // td1[31:16]=0 | tile_dim0=72
      g1.s4 = (int)(unsigned)AH;          // tile_dim1=68 | tile_dim2=0
      g1.s5 = W;                          // tensor_dim0_stride[31:0]
      g1.s6 = 0;                          // stride0[47:32] | stride1[15:0]
      g1.s7 = 0;                          // stride1[47:16]

      i32x4_t gz4 = {0, 0, 0, 0};                // groups 2/3 unused (2D)
      i32x8_t gz8 = {0, 0, 0, 0, 0, 0, 0, 0};    // trailing group (unused)
      __builtin_amdgcn_tensor_load_to_lds(g0, g1, gz4, gz4, gz8, 0);
      wait_tensor();
    }
#else
    // Fallback fast path: aligned float4 segments, async b128 into LDS.
    for (int i = tid; i < AH * NSEG; i += NTHREADS) {
      const int row = i / NSEG;
      const int seg = i - row * NSEG;
      const float* gp =
          img + (size_t)(by - HALOY + row) * (size_t)W + (bx - XOFF + 4 * seg);
      float* lp = &a[row * AWS + 4 * seg];
#if defined(CH_ASYNC128)
      __builtin_amdgcn_global_load_async_to_lds_b128(
          (g1_v4i_t*)gp, (l3_v4i_t*)lp, /*offset=*/0, /*cpol=*/0);
#else
      *(float4*)lp = *(const float4*)gp;
#endif
    }
    wait_async_lds();
#endif
  } else {
    // Border path: per-element edge clamping.
    for (int i = tid; i < AH * AWS; i += NTHREADS) {
      const int ly = i / AWS;
      const int lx = i - ly * AWS;
      int gy = by + ly - HALOY;
      int gx = bx + lx - XOFF;
      gy = min(max(gy, 0), H - 1);
      gx = min(max(gx, 0), W - 1);
      const size_t goff = (size_t)gy * (size_t)W + (size_t)gx;
#if defined(CH_ASYNC32)
      __builtin_amdgcn_global_load_async_to_lds_b32(
          (g1_int_t*)(img + goff), (l3_int_t*)&a[i], /*offset=*/0, /*cpol=*/0);
#else
      a[i] = img[goff];
#endif
    }
    wait_async_lds();
  }
  __syncthreads();

  // ---- Stage 2: x1 = -lap(x) - x + x^3 on 66x66 (halo 1), with the x1 halo
  // evaluated at edge-clamped CENTER coordinates (== replication pad of x1).
  for (int i = tid; i < BH * BW; i += NTHREADS) {
    const int ly = i / BW;
    const int lx = i - ly * BW;
    const int gy = by + ly - 1;
    const int gx = bx + lx - 1;
    const int cy = min(max(gy, 0), H - 1); // clamp center -> x1 replication
    const int cx = min(max(gx, 0), W - 1);
    const int ay = cy - by + HALOY; // rows: in [1, TY+2] subset of [0, AH-1]
    const int ax = cx - bx + XOFF;  // cols: in [3, TX+4] subset of [0, AWS-1]
    const float c = a[ay * AWS + ax];
    const float lap = a[(ay - 1) * AWS + ax] + a[(ay + 1) * AWS + ax] +
                      a[ay * AWS + (ax - 1)] + a[ay * AWS + (ax + 1)] -
                      4.0f * c;
    b[i] = fmaf(c * c, c, -(lap + c)); // x1 = c^3 - c - lap
  }
  __syncthreads();

  // ---- Stage 3: out = x + 0.1 * lap(x1); 4 outputs/iter, b128 stores ----
  for (int i = tid; i < TY * OSEG; i += NTHREADS) {
    const int row = i / OSEG;
    const int seg = i - row * OSEG;
    const int col0 = 4 * seg;

    const int mbase = (row + 1) * BW + col0; // x1 center row, col0 = left nbr
    const float m0 = b[mbase + 0];
    const float m1 = b[mbase + 1];
    const float m2 = b[mbase + 2];
    const float m3 = b[mbase + 3];
    const float m4 = b[mbase + 4];
    const float m5 = b[mbase + 5];
    const int ubase = row * BW + col0 + 1;
    const float u0 = b[ubase + 0], u1 = b[ubase + 1];
    const float u2 = b[ubase + 2], u3 = b[ubase + 3];
    const int dbase = (row + 2) * BW + col0 + 1;
    const float d0 = b[dbase + 0], d1 = b[dbase + 1];
    const float d2 = b[dbase + 2], d3 = b[dbase + 3];

    // x centers: 16B-aligned float4 in LDS (row stride 72, col0 multiple of 4)
    const float4 xv = *(const float4*)&a[(row + HALOY) * AWS + (col0 + XOFF)];

    float4 o;
    o.x = fmaf(0.1f, u0 + d0 + m0 + m2 - 4.0f * m1, xv.x);
    o.y = fmaf(0.1f, u1 + d1 + m1 + m3 - 4.0f * m2, xv.y);
    o.z = fmaf(0.1f, u2 + d2 + m2 + m4 - 4.0f * m3, xv.z);
    o.w = fmaf(0.1f, u3 + d3 + m3 + m5 - 4.0f * m4, xv.w);

    const int oy = by + row;
    const int ox = bx + col0;
    if (oy < H && ox + 3 < W) {
      *(float4*)(oimg + (size_t)oy * (size_t)W + (size_t)ox) = o; // b128 store
    }
  }
}

extern "C" void kernel_launch(void* const* d_in, const int* in_sizes, int n_in,
                              void* d_out, int out_size, void* d_ws,
                              size_t ws_size, hipStream_t stream) {
  (void)n_in;
  (void)out_size;
  (void)d_ws;
  (void)ws_size;
  const float* x = (const float*)d_in[0];
  float* out = (float*)d_out;

  const int W = 2048, H = 2048;
  const int B = in_sizes[0] / (W * H); // 8 images, 1 channel
  dim3 grid((W + TX - 1) / TX, (H + TY - 1) / TY, B);
  ch_step_fused<<<grid, dim3(NTHREADS), 0, stream>>>(x, out, W, H);
}